// SPAIRGlimpseZPresMLP_64269890617423
// MI455X (gfx1250) — compile-verified
//
#include <hip/hip_runtime.h>

typedef __attribute__((ext_vector_type(2))) float v2f;
typedef __attribute__((ext_vector_type(8))) float v8f;

#define EPSV 1e-6f

// ---------------------------------------------------------------------------
// Kernel 0: zero the accumulators (segmax bits / segment sums / member_center)
// ---------------------------------------------------------------------------
__global__ void spair_init_kernel(unsigned* __restrict__ segmax_bits,
                                  float* __restrict__ segsum,
                                  float* __restrict__ center, int G) {
  int i = blockIdx.x * blockDim.x + threadIdx.x;
  if (i < G) {
    segmax_bits[i] = 0u;   // encodes below any real float's encoding
    segsum[i] = 0.0f;
  }
  if (i < 3 * G) center[i] = 0.0f;
}

// Monotone float->uint order-preserving encoding for atomicMax
__device__ __forceinline__ unsigned enc_f32(float f) {
  unsigned u = __float_as_uint(f);
  return (u & 0x80000000u) ? ~u : (u | 0x80000000u);
}
__device__ __forceinline__ float dec_f32(unsigned u) {
  unsigned bits = (u & 0x80000000u) ? (u & 0x7FFFFFFFu) : ~u;
  return __uint_as_float(bits);
}

// ---------------------------------------------------------------------------
// Kernel 1: per-segment max of log_mask (global_atomic_max_u32)
// ---------------------------------------------------------------------------
__global__ void spair_segmax_kernel(const float* __restrict__ lm,
                                    const int* __restrict__ idx,
                                    unsigned* __restrict__ segmax_bits, int M) {
  int i = blockIdx.x * blockDim.x + threadIdx.x;
  if (i >= M) return;
  atomicMax(&segmax_bits[idx[i]], enc_f32(lm[i]));
}

// ---------------------------------------------------------------------------
// Kernel 2: fused pass — w = exp(lm - segmax); atomically accumulate
//           Σw per segment and Σ(w * pos) into member_center (unnormalized)
// ---------------------------------------------------------------------------
__global__ void spair_accum_kernel(const float* __restrict__ lm,
                                   const int* __restrict__ idx,
                                   const float* __restrict__ pos,
                                   const unsigned* __restrict__ segmax_bits,
                                   float* __restrict__ segsum,
                                   float* __restrict__ center, int M) {
  int i = blockIdx.x * blockDim.x + threadIdx.x;
  if (i >= M) return;
  int g = idx[i];
  float mx = dec_f32(segmax_bits[g]);
  float w = __expf(lm[i] - mx);
  float px = pos[3 * (size_t)i + 0];
  float py = pos[3 * (size_t)i + 1];
  float pz = pos[3 * (size_t)i + 2];
  atomicAdd(&segsum[g], w);
  atomicAdd(&center[3 * (size_t)g + 0], w * px);
  atomicAdd(&center[3 * (size_t)g + 1], w * py);
  atomicAdd(&center[3 * (size_t)g + 2], w * pz);
}

// ---------------------------------------------------------------------------
// Kernel 3: normalize member_center by segment sum (empty segment -> 0)
// ---------------------------------------------------------------------------
__global__ void spair_normalize_kernel(const float* __restrict__ segsum,
                                       float* __restrict__ center, int G) {
  int i = blockIdx.x * blockDim.x + threadIdx.x;
  if (i >= G) return;
  float s = segsum[i];
  float inv = (s > 0.0f) ? (1.0f / s) : 0.0f;
  center[3 * (size_t)i + 0] *= inv;
  center[3 * (size_t)i + 1] *= inv;
  center[3 * (size_t)i + 2] *= inv;
}

// ---------------------------------------------------------------------------
// Kernel 4: z_pres head via V_WMMA_F32_16X16X4_F32.
// One wave handles 16 glimpse rows; K loop over 256 in steps of 4.
// A layout (ISA 7.12.2, 32-bit A 16x4): lane L<16 holds A[L, k0..k0+1],
// lane L>=16 holds A[L-16, k0+2..k0+3]  -> per-lane float2 load.
// B columns all replicate W's K-chunk   -> same per-lane float2 addressing.
// D: vgpr r, lanes 0-15 = row r col lane; lanes 16-31 = row r+8.
// All columns identical; lanes 0/16 run the scalar epilogue for 8 rows each.
// ---------------------------------------------------------------------------
__global__ void __launch_bounds__(256)
spair_head_kernel(const float* __restrict__ feat, const float* __restrict__ W,
                  const float* __restrict__ bptr, const float* __restrict__ tptr,
                  const float* __restrict__ noise, float* __restrict__ out, int G) {
  const int lane = threadIdx.x & 31;
  const int wave = threadIdx.x >> 5;
  const int tile = blockIdx.x * 8 + wave;     // 16-row tile id
  const int row_base = tile * 16;
  if (row_base >= G) return;                  // wave-uniform; EXEC stays full

  const int krow = lane & 15;                 // M index within tile
  const int khalf = (lane >> 4) << 1;         // 0 (lanes 0-15) or 2 (16-31)
  const float* arow = feat + (size_t)(row_base + krow) * 256 + khalf;
  const float* wrow = W + khalf;

  v8f acc = {};
#pragma unroll 8
  for (int k0 = 0; k0 < 256; k0 += 4) {
    v2f a = *(const v2f*)(arow + k0);
    v2f bm = *(const v2f*)(wrow + k0);
    acc = __builtin_amdgcn_wmma_f32_16x16x4_f32(
        /*neg_a=*/false, a, /*neg_b=*/false, bm,
        /*c_mod=*/(short)0, acc, /*reuse_a=*/false, /*reuse_b=*/false);
  }

  if ((lane & 15) == 0) {
    const float bias = bptr[0];
    const float invt = 1.0f / tptr[0];
    const int mbase = row_base + ((lane >> 4) << 3);  // lane0: rows 0-7, lane16: 8-15
#pragma unroll
    for (int r = 0; r < 8; ++r) {
      int row = mbase + r;
      float d = acc[r];
      float logit = 8.8f * tanhf(d + bias);
      float u = noise[row];
      u = fminf(fmaxf(u, EPSV), 1.0f - EPSV);
      float rl = (logit + __logf(u) - log1pf(-u)) * invt;
      // log_sigmoid, numerically stable
      float lzp = (rl >= 0.0f) ? -log1pf(__expf(-rl)) : (rl - log1pf(__expf(rl)));
      out[row] = lzp;          // log_z_pres
      out[G + row] = logit;    // logit_pres
    }
  }
}

extern "C" void kernel_launch(void* const* d_in, const int* in_sizes, int n_in,
                              void* d_out, int out_size, void* d_ws, size_t ws_size,
                              hipStream_t stream) {
  const float* feat  = (const float*)d_in[0];  // [G,256]
  const float* pos   = (const float*)d_in[1];  // [M,3]
  const float* lm    = (const float*)d_in[2];  // [M,1]
  const int*   idx   = (const int*)d_in[3];    // [M]
  const float* temp  = (const float*)d_in[4];  // scalar
  const float* noise = (const float*)d_in[5];  // [G]
  const float* W     = (const float*)d_in[6];  // [1,256]
  const float* b     = (const float*)d_in[7];  // [1]

  const int M = in_sizes[3];
  const int G = in_sizes[5];

  float* out = (float*)d_out;
  float* center = out + 2 * (size_t)G;                 // [G,3] slice of d_out

  unsigned* segmax_bits = (unsigned*)d_ws;             // G u32
  float* segsum = (float*)((char*)d_ws + (size_t)G * sizeof(unsigned));  // G f32

  const int B = 256;
  spair_init_kernel<<<(3 * G + B - 1) / B, B, 0, stream>>>(segmax_bits, segsum,
                                                           center, G);
  spair_segmax_kernel<<<(M + B - 1) / B, B, 0, stream>>>(lm, idx, segmax_bits, M);
  spair_accum_kernel<<<(M + B - 1) / B, B, 0, stream>>>(lm, idx, pos, segmax_bits,
                                                        segsum, center, M);
  spair_normalize_kernel<<<(G + B - 1) / B, B, 0, stream>>>(segsum, center, G);

  const int tiles = (G + 15) / 16;
  spair_head_kernel<<<(tiles + 7) / 8, B, 0, stream>>>(feat, W, b, temp, noise,
                                                       out, G);
}